// GNCA_81140522156681
// MI455X (gfx1250) — compile-verified
//
#include <hip/hip_runtime.h>
#include <math.h>

typedef __attribute__((ext_vector_type(2))) float v2f;
typedef __attribute__((ext_vector_type(8))) float v8f;

// ---------- float ordered-key helpers for atomic segment-max ----------
__device__ __forceinline__ unsigned fkey(float x) {
    unsigned u = __float_as_uint(x);
    return (u & 0x80000000u) ? ~u : (u | 0x80000000u);
}
__device__ __forceinline__ float funkey(unsigned k) {
    unsigned u = (k & 0x80000000u) ? (k ^ 0x80000000u) : ~k;
    return __uint_as_float(u);
}
#define NEG_INF_KEY 0x007FFFFFu   // fkey(-inf)

// ---------- fp32 WMMA 16x16x4 tile: Y16x16 = X[base..base+15, 0..4] @ Bp(8x16) ----------
// A layout (ISA 7.12.2, 32-bit A 16x4): lanes 0-15 hold K=0(V0),K=1(V1); lanes 16-31 K=2,K=3.
// B layout mirrored: lanes 0-15 hold K rows {0,1}; lanes 16-31 rows {2,3} per 2-VGPR operand.
// D layout: VGPR r: lanes 0-15 -> (M=r, N=lane), lanes 16-31 -> (M=r+8, N=lane-16).
__device__ __forceinline__ v8f gemm5_tile(const float* __restrict__ X, int base,
                                          int stride, const float* __restrict__ Bp, int N) {
    int lane = threadIdx.x & 31;
    int hsel = lane >> 4;
    int n    = lane & 15;
    int r    = base + n;
    if (r > N - 1) r = N - 1;
    const float* xr = X + r * stride;
    float x0 = xr[0], x1 = xr[1], x2 = xr[2], x3 = xr[3], x4 = xr[4];
    v2f a0, a1, b0, b1;
    a0[0] = hsel ? x2  : x0;
    a0[1] = hsel ? x3  : x1;
    a1[0] = hsel ? 0.f : x4;
    a1[1] = 0.f;
    b0[0] = Bp[(2 * hsel + 0) * 16 + n];   // K rows 0/2
    b0[1] = Bp[(2 * hsel + 1) * 16 + n];   // K rows 1/3
    b1[0] = Bp[(2 * hsel + 4) * 16 + n];   // K rows 4/6 (5..7 zero padding)
    b1[1] = Bp[(2 * hsel + 5) * 16 + n];
    v8f c = {};
    c = __builtin_amdgcn_wmma_f32_16x16x4_f32(false, a0, false, b0, (short)0, c, false, false);
    c = __builtin_amdgcn_wmma_f32_16x16x4_f32(false, a1, false, b1, (short)0, c, false, false);
    return c;
}

// ---------- prep: pack B matrices, fold biases, collapse edge projection ----------
__global__ void k_prep(const float* W, const float* att_src, const float* att_dst,
                       const float* We, const float* att_edge, const float* gat_bias,
                       const float* W1, const float* b1, const float* W2, const float* b2,
                       float* BpA, float* Bp1, float* Bp2,
                       float* bias1_16, float* bias2_16, float* coef) {
    if (blockIdx.x || threadIdx.x) return;
    for (int i = 0; i < 128; ++i) { BpA[i] = 0.f; Bp1[i] = 0.f; Bp2[i] = 0.f; }
    for (int i = 0; i < 16; ++i)  { bias1_16[i] = 0.f; bias2_16[i] = 0.f; }
    for (int k = 0; k < 5; ++k) {
        for (int nn = 0; nn < 5; ++nn) BpA[k * 16 + nn] = W[k * 5 + nn];
        float s = 0.f, d = 0.f;
        for (int j = 0; j < 5; ++j) { s += W[k * 5 + j] * att_src[j]; d += W[k * 5 + j] * att_dst[j]; }
        BpA[k * 16 + 5] = s;   // column 5 -> a_src
        BpA[k * 16 + 6] = d;   // column 6 -> a_dst
        for (int nn = 0; nn < 5; ++nn) Bp1[k * 16 + nn] = W1[k * 5 + nn];
        for (int nn = 0; nn < 2; ++nn) Bp2[k * 16 + nn] = W2[k * 2 + nn];
    }
    for (int nn = 0; nn < 5; ++nn) {
        float s = b1[nn];
        for (int j = 0; j < 5; ++j) s += gat_bias[j] * W1[j * 5 + nn];  // fold gat_bias
        bias1_16[nn] = s;
    }
    bias2_16[0] = b2[0]; bias2_16[1] = b2[1];
    float w0 = 0.f, w1 = 0.f;
    for (int j = 0; j < 5; ++j) { w0 += We[j] * att_edge[j]; w1 += We[5 + j] * att_edge[j]; }
    coef[0] = w0; coef[1] = w1;  // a_e = dist*w0 + cell*w1
}

// ---------- init node-side accumulators + scalar output slots ----------
__global__ void k_init(unsigned* mkey, float* denom, float* outg,
                       int* closec, int* cellc, float* scal, int N) {
    int i = blockIdx.x * blockDim.x + threadIdx.x;
    if (i < N) {
        mkey[i]  = NEG_INF_KEY;
        denom[i] = 0.f;
        closec[i] = 0; cellc[i] = 0;
        float* o = outg + i * 8;
        o[0] = 0.f; o[1] = 0.f; o[2] = 0.f; o[3] = 0.f; o[4] = 0.f;
    }
    if (i < 5) scal[i] = 0.f;
}

// ---------- WMMA stage 1: hsb[.,0:5)=x@W, hsb[.,5]=a_src, hsb[.,6]=a_dst ----------
__global__ void k_hs(const float* __restrict__ x, const float* __restrict__ BpA,
                     float* __restrict__ hsb, int ntiles, int N) {
    int nw   = gridDim.x * (blockDim.x >> 5);
    int wid  = blockIdx.x * (blockDim.x >> 5) + (threadIdx.x >> 5);
    int hsel = (threadIdx.x >> 4) & 1;
    int n    = threadIdx.x & 15;
    for (int t = wid; t < ntiles; t += nw) {
        int base = t * 16;
        v8f c = gemm5_tile(x, base, 5, BpA, N);
        int row0 = base + hsel * 8;
        if (base + 16 <= N) {                 // wave-uniform full-tile fast path
            if (n < 7) {
#pragma unroll
                for (int r = 0; r < 8; ++r) hsb[(row0 + r) * 8 + n] = c[r];
            }
        } else {
#pragma unroll
            for (int r = 0; r < 8; ++r)
                if (n < 7 && row0 + r < N) hsb[(row0 + r) * 8 + n] = c[r];
        }
    }
}

// ---------- WMMA MLP layer: Y = relu(X@Bp + bias), X stride 8 ----------
__global__ void k_mlp(const float* __restrict__ X, const float* __restrict__ Bp,
                      const float* __restrict__ bias16, float* __restrict__ Y,
                      int ystride, int ncols_out, int ntiles, int N) {
    int nw   = gridDim.x * (blockDim.x >> 5);
    int wid  = blockIdx.x * (blockDim.x >> 5) + (threadIdx.x >> 5);
    int hsel = (threadIdx.x >> 4) & 1;
    int n    = threadIdx.x & 15;
    float bias = bias16[n];
    for (int t = wid; t < ntiles; t += nw) {
        int base = t * 16;
        v8f c = gemm5_tile(X, base, 8, Bp, N);
        int row0 = base + hsel * 8;
        if (base + 16 <= N) {                 // wave-uniform full-tile fast path
            if (n < ncols_out) {
#pragma unroll
                for (int r = 0; r < 8; ++r) {
                    float v = c[r] + bias;
                    Y[(row0 + r) * ystride + n] = v > 0.f ? v : 0.f;
                }
            }
        } else {
#pragma unroll
            for (int r = 0; r < 8; ++r) {
                float v = c[r] + bias;
                if (n < ncols_out && row0 + r < N) Y[(row0 + r) * ystride + n] = v > 0.f ? v : 0.f;
            }
        }
    }
}

// ---------- shared logit recompute (a_src/a_dst packed in hsb cols 5/6) ----------
__device__ __forceinline__ float edge_logit(const int* ei, const float* ea, const float* hsb,
                                            float w0, float w1, int e, int E,
                                            int& s, int& d, float& dist, float& cell) {
    s = ei[e]; d = ei[E + e];
    dist = ea[2 * e]; cell = ea[2 * e + 1];
    float l = hsb[s * 8 + 5] + hsb[d * 8 + 6] + dist * w0 + cell * w1;
    return l >= 0.f ? l : 0.2f * l;   // leaky_relu 0.2
}

// ---------- edge pass 1: segment-max of logits over dst + fused src counts ----------
__global__ void k_edge1(const int* __restrict__ ei, const float* __restrict__ ea,
                        const float* __restrict__ hsb, const float* __restrict__ coef,
                        unsigned* __restrict__ mkey, int* __restrict__ closec,
                        int* __restrict__ cellc, int E) {
    int e = blockIdx.x * blockDim.x + threadIdx.x;
    if (e >= E) return;
    int s, d; float dist, cell;
    float l = edge_logit(ei, ea, hsb, coef[0], coef[1], e, E, s, d, dist, cell);
    atomicMax(&mkey[d], fkey(l));
    if (dist < 0.1f)  atomicAdd(&closec[s], 1);   // CONSUME_RADIUS = 0.1
    if (cell == 1.0f) atomicAdd(&cellc[s], 1);
}

// ---------- edge pass 2: denom = segment-sum of exp(logit - m) ----------
__global__ void k_edge2(const int* __restrict__ ei, const float* __restrict__ ea,
                        const float* __restrict__ hsb, const float* __restrict__ coef,
                        const unsigned* __restrict__ mkey, float* __restrict__ denom, int E) {
    int e = blockIdx.x * blockDim.x + threadIdx.x;
    if (e >= E) return;
    int s, d; float dist, cell;
    float l = edge_logit(ei, ea, hsb, coef[0], coef[1], e, E, s, d, dist, cell);
    atomicAdd(&denom[d], __expf(l - funkey(mkey[d])));
}

// ---------- edge pass 3: out[dst] += hs[src] * alpha ----------
__global__ void k_edge3(const int* __restrict__ ei, const float* __restrict__ ea,
                        const float* __restrict__ hsb, const float* __restrict__ coef,
                        const unsigned* __restrict__ mkey, const float* __restrict__ denom,
                        float* __restrict__ outg, int E) {
    int e = blockIdx.x * blockDim.x + threadIdx.x;
    if (e >= E) return;
    int s, d; float dist, cell;
    float l = edge_logit(ei, ea, hsb, coef[0], coef[1], e, E, s, d, dist, cell);
    float ev  = __expf(l - funkey(mkey[d]));
    float den = denom[d];
    float alpha = ev / (den > 1e-16f ? den : 1e-16f);
    const float4 hv = *(const float4*)(hsb + s * 8);   // 32B-aligned b128
    float h4 = hsb[s * 8 + 4];
    float* orow = outg + d * 8;
    atomicAdd(&orow[0], hv.x * alpha);
    atomicAdd(&orow[1], hv.y * alpha);
    atomicAdd(&orow[2], hv.z * alpha);
    atomicAdd(&orow[3], hv.w * alpha);
    atomicAdd(&orow[4], h4  * alpha);
}

// ---------- final node update + scalar reductions ----------
__global__ void k_final(const float* __restrict__ x, const float* __restrict__ u,
                        const int* __restrict__ closec, const int* __restrict__ cellc,
                        float* __restrict__ newx, float* __restrict__ scal,
                        float invN, int N) {
    __shared__ float sred[256];
    int i = blockIdx.x * blockDim.x + threadIdx.x;
    float vals[5] = {0.f, 0.f, 0.f, 0.f, 0.f};
    if (i < N) {
        const float* xr = x + i * 5;
        float h0 = u[i * 2 + 0] * 2.f - 1.f;
        float h1 = u[i * 2 + 1] * 2.f - 1.f;
        float isc = (xr[4] == 1.f) ? 1.f : 0.f;
        float v0 = xr[2] + h0 * 0.01f * isc;
        float v1 = xr[3] + h1 * 0.01f * isc;
        v0 = fminf(fmaxf(v0, -0.1f), 0.1f);
        v1 = fminf(fmaxf(v1, -0.1f), 0.1f);
        float p0 = xr[0] + v0, p1 = xr[1] + v1;
        float* nr = newx + i * 5;
        nr[0] = p0; nr[1] = p1; nr[2] = v0; nr[3] = v1; nr[4] = xr[4];
        vals[0] = fabsf(v0) * invN;
        vals[1] = fabsf(v1) * invN;
        float bc = 0.f;
        if (fabsf(p0) > 1.f) bc += logf(fabsf(p0) + 1e-6f);
        if (fabsf(p1) > 1.f) bc += logf(fabsf(p1) + 1e-6f);
        vals[2] = bc;
        vals[3] = (xr[4] == 0.f && closec[i] >= 3) ? 1.f : 0.f;  // consumed food
        vals[4] = (xr[4] == 1.f && cellc[i] < 1) ? 1.f : 0.f;    // dead island cells
    }
    for (int j = 0; j < 5; ++j) {
        sred[threadIdx.x] = vals[j];
        __syncthreads();
        for (int s = 128; s > 0; s >>= 1) {
            if (threadIdx.x < (unsigned)s) sred[threadIdx.x] += sred[threadIdx.x + s];
            __syncthreads();
        }
        if (threadIdx.x == 0) atomicAdd(&scal[j], sred[0]);
        __syncthreads();
    }
}

extern "C" void kernel_launch(void* const* d_in, const int* in_sizes, int n_in,
                              void* d_out, int out_size, void* d_ws, size_t ws_size,
                              hipStream_t stream) {
    const float* x        = (const float*)d_in[0];
    const int*   ei       = (const int*)  d_in[1];
    const float* ea       = (const float*)d_in[2];
    const float* W        = (const float*)d_in[3];
    const float* att_src  = (const float*)d_in[4];
    const float* att_dst  = (const float*)d_in[5];
    const float* We       = (const float*)d_in[6];
    const float* att_edge = (const float*)d_in[7];
    const float* gat_bias = (const float*)d_in[8];
    const float* W1       = (const float*)d_in[9];
    const float* b1       = (const float*)d_in[10];
    const float* W2       = (const float*)d_in[11];
    const float* b2       = (const float*)d_in[12];

    const int N = in_sizes[0] / 5;
    const int E = in_sizes[1] / 2;

    // workspace carve-up (64B-aligned chunks)
    float* ws = (float*)d_ws;
    size_t off = 0;
    auto alloc = [&](size_t nelem) { float* p = ws + off; off += (nelem + 15) & ~(size_t)15; return p; };
    float*    hsb      = alloc((size_t)N * 8);   // cols 0-4: hs, 5: a_src, 6: a_dst
    unsigned* mkey     = (unsigned*)alloc(N);
    float*    denom    = alloc(N);
    float*    outg     = alloc((size_t)N * 8);   // GAT output, stride 8
    float*    tbuf     = alloc((size_t)N * 8);   // MLP layer-1 output, stride 8
    float*    ubuf     = alloc((size_t)N * 2);   // MLP layer-2 output
    int*      closec   = (int*)alloc(N);
    int*      cellc    = (int*)alloc(N);
    float*    BpA      = alloc(128);
    float*    Bp1      = alloc(128);
    float*    Bp2      = alloc(128);
    float*    bias1_16 = alloc(16);
    float*    bias2_16 = alloc(16);
    float*    coef     = alloc(16);

    float* d_outf = (float*)d_out;
    float* newx   = d_outf;
    float* scal   = d_outf + (size_t)N * 5;   // [vmean0, vmean1, border, consumed, dead]

    const int ntiles = (N + 15) / 16;
    const int nblk   = (N + 255) / 256;
    const int eblk   = (E + 255) / 256;

    k_prep<<<1, 1, 0, stream>>>(W, att_src, att_dst, We, att_edge, gat_bias,
                                W1, b1, W2, b2, BpA, Bp1, Bp2, bias1_16, bias2_16, coef);
    k_init<<<nblk, 256, 0, stream>>>(mkey, denom, outg, closec, cellc, scal, N);
    k_hs<<<256, 256, 0, stream>>>(x, BpA, hsb, ntiles, N);
    k_edge1<<<eblk, 256, 0, stream>>>(ei, ea, hsb, coef, mkey, closec, cellc, E);
    k_edge2<<<eblk, 256, 0, stream>>>(ei, ea, hsb, coef, mkey, denom, E);
    k_edge3<<<eblk, 256, 0, stream>>>(ei, ea, hsb, coef, mkey, denom, outg, E);
    k_mlp<<<256, 256, 0, stream>>>(outg, Bp1, bias1_16, tbuf, 8, 5, ntiles, N);
    k_mlp<<<256, 256, 0, stream>>>(tbuf, Bp2, bias2_16, ubuf, 2, 2, ntiles, N);
    k_final<<<nblk, 256, 0, stream>>>(x, ubuf, closec, cellc, newx, scal, 1.0f / (float)N, N);
}